// DAWN_47699906789385
// MI455X (gfx1250) — compile-verified
//
#include <hip/hip_runtime.h>
#include <math.h>

#define D_MODEL 512
#define RANK 128
#define N_HEADS 8
#define D_HEAD 16
#define N_COMPRESS 64
#define N_EXPAND 64
#define N_KNOW 32768
#define KNOW_K 8
#define N_LAYERS 2
#define BATCH 2
#define SEQ 512
#define NTOK (BATCH * SEQ)

#define CHUNK 512
#define NSPLIT 8                            // grid-Y partitions of the N_KNOW axis
#define NCHUNKS (N_KNOW / CHUNK)            // 64
#define CHPS (NCHUNKS / NSPLIT)             // 8 chunks per split
#define NCAND (NSPLIT * 8)                  // 64 candidates per token

typedef float v2f __attribute__((ext_vector_type(2)));
typedef float v8f __attribute__((ext_vector_type(8)));

__device__ __forceinline__ v8f wmma_f32_16x16x4(v2f a, v2f b, v8f c) {
  // D = A(16x4 f32) * B(4x16 f32) + C(16x16 f32)
  return __builtin_amdgcn_wmma_f32_16x16x4_f32(false, a, false, b, (short)0, c,
                                               false, false);
}

// ---------------------------------------------------------------- LayerNorm
__global__ void ln_kernel(const float* __restrict__ x, const float* __restrict__ g,
                          const float* __restrict__ b, float* __restrict__ out) {
  int tok = blockIdx.x;
  int tid = threadIdx.x;  // 256 threads
  const float* xp = x + (size_t)tok * D_MODEL;
  float v0 = xp[tid], v1 = xp[tid + 256];
  __shared__ float rs[256], rq[256];
  rs[tid] = v0 + v1;
  rq[tid] = v0 * v0 + v1 * v1;
  __syncthreads();
  for (int s = 128; s > 0; s >>= 1) {
    if (tid < s) { rs[tid] += rs[tid + s]; rq[tid] += rq[tid + s]; }
    __syncthreads();
  }
  float mean = rs[0] * (1.0f / D_MODEL);
  float var = rq[0] * (1.0f / D_MODEL) - mean * mean;
  float rstd = rsqrtf(var + 1e-5f);
  float* op = out + (size_t)tok * D_MODEL;
  op[tid] = (v0 - mean) * rstd * g[tid] + b[tid];
  op[tid + 256] = (v1 - mean) * rstd * g[tid + 256] + b[tid + 256];
}

// ------------------------------------------------- routed compress (D -> R)
__global__ void compress_kernel(const float* __restrict__ h,
                                const float* __restrict__ router,   // [64,512]
                                const float* __restrict__ neurons,  // [64,512,128]
                                float* __restrict__ out) {          // [NTOK,128]
  int tok = blockIdx.x;
  int tid = threadIdx.x;  // 128 threads
  __shared__ float hs[D_MODEL];
  __shared__ float sc[N_COMPRESS];
  __shared__ float sw[2];
  __shared__ int se[2];
  const float* hp = h + (size_t)tok * D_MODEL;
  for (int i = tid; i < D_MODEL; i += 128) hs[i] = hp[i];
  __syncthreads();
  if (tid < N_COMPRESS) {
    const float* rp = router + (size_t)tid * D_MODEL;
    float acc = 0.f;
    for (int d = 0; d < D_MODEL; ++d) acc += hs[d] * rp[d];
    sc[tid] = acc;
  }
  __syncthreads();
  if (tid == 0) {
    float s1 = -INFINITY, s2 = -INFINITY;
    int e1 = 0, e2 = 0;
    for (int e = 0; e < N_COMPRESS; ++e) {
      float s = sc[e];
      if (s > s1) { s2 = s1; e2 = e1; s1 = s; e1 = e; }
      else if (s > s2) { s2 = s; e2 = e; }
    }
    float z = expf(s2 - s1);
    sw[0] = 1.f / (1.f + z);
    sw[1] = z / (1.f + z);
    se[0] = e1; se[1] = e2;
  }
  __syncthreads();
  int e0 = se[0], e1i = se[1];
  float w0 = sw[0], w1 = sw[1];
  const float* n0 = neurons + (size_t)e0 * (D_MODEL * RANK) + tid;
  const float* n1 = neurons + (size_t)e1i * (D_MODEL * RANK) + tid;
  float a0 = 0.f, a1 = 0.f;
  for (int d = 0; d < D_MODEL; ++d) {
    float hv = hs[d];
    a0 += hv * n0[(size_t)d * RANK];   // coalesced: consecutive tid -> consecutive r
    a1 += hv * n1[(size_t)d * RANK];
  }
  out[(size_t)tok * RANK + tid] = w0 * a0 + w1 * a1;
}

// --------------------------------------------------- routed expand (R -> D)
__global__ void expand_kernel(const float* __restrict__ in,        // [NTOK,128]
                              const float* __restrict__ router,    // [64,128]
                              const float* __restrict__ neurons,   // [64,128,512]
                              float* __restrict__ x) {             // += [NTOK,512]
  int tok = blockIdx.x;
  int tid = threadIdx.x;  // 256 threads
  __shared__ float is[RANK];
  __shared__ float sc[N_EXPAND];
  __shared__ float sw[2];
  __shared__ int se[2];
  const float* ip = in + (size_t)tok * RANK;
  if (tid < RANK) is[tid] = ip[tid];
  __syncthreads();
  if (tid < N_EXPAND) {
    const float* rp = router + (size_t)tid * RANK;
    float acc = 0.f;
    for (int r = 0; r < RANK; ++r) acc += is[r] * rp[r];
    sc[tid] = acc;
  }
  __syncthreads();
  if (tid == 0) {
    float s1 = -INFINITY, s2 = -INFINITY;
    int e1 = 0, e2 = 0;
    for (int e = 0; e < N_EXPAND; ++e) {
      float s = sc[e];
      if (s > s1) { s2 = s1; e2 = e1; s1 = s; e1 = e; }
      else if (s > s2) { s2 = s; e2 = e; }
    }
    float z = expf(s2 - s1);
    sw[0] = 1.f / (1.f + z);
    sw[1] = z / (1.f + z);
    se[0] = e1; se[1] = e2;
  }
  __syncthreads();
  int e0 = se[0], e1i = se[1];
  float w0 = sw[0], w1 = sw[1];
  for (int d = tid; d < D_MODEL; d += 256) {
    const float* n0 = neurons + (size_t)e0 * (RANK * D_MODEL) + d;
    const float* n1 = neurons + (size_t)e1i * (RANK * D_MODEL) + d;
    float a0 = 0.f, a1 = 0.f;
    for (int r = 0; r < RANK; ++r) {
      float iv = is[r];
      a0 += iv * n0[(size_t)r * D_MODEL];
      a1 += iv * n1[(size_t)r * D_MODEL];
    }
    x[(size_t)tok * D_MODEL + d] += w0 * a0 + w1 * a1;
  }
}

// -------------------------- flash attention: one wave per (b, head, q-tile)
// WMMA f32 16x16x4: QK^T (K=16 -> 4 wmma) and P@V (K=16 -> 4 wmma) per tile.
__global__ void attn_kernel(const float* __restrict__ Q, const float* __restrict__ K,
                            const float* __restrict__ V, float* __restrict__ O) {
  int bid = blockIdx.x;
  int qt = bid % (SEQ / 16);
  int head = (bid / (SEQ / 16)) % N_HEADS;
  int b = bid / ((SEQ / 16) * N_HEADS);
  int lane = threadIdx.x;  // block = 32 = one wave
  int nl = lane & 15;
  int kh = lane >> 4;      // which K half (A/B layout: lanes 16-31 hold K+2)
  int mrow = kh ? 8 : 0;   // C/D layout: lanes 16-31 hold rows 8-15
  size_t base = (size_t)b * SEQ;
  __shared__ float Pt[16][17];

  // A fragments for Q tile (16 rows x K=16), 4 chunks of K=4
  const float* qp = Q + (base + qt * 16 + nl) * RANK + head * D_HEAD;
  v2f aq[4];
#pragma unroll
  for (int kc = 0; kc < 4; ++kc) aq[kc] = *(const v2f*)(qp + 4 * kc + 2 * kh);

  v8f o;
  float rm[8], rl[8];
#pragma unroll
  for (int v = 0; v < 8; ++v) { o[v] = 0.f; rm[v] = -INFINITY; rl[v] = 0.f; }

  for (int j = 0; j <= qt; ++j) {
    v8f s;
#pragma unroll
    for (int v = 0; v < 8; ++v) s[v] = 0.f;
    const float* kp = K + (base + j * 16 + nl) * RANK + head * D_HEAD;
#pragma unroll
    for (int kc = 0; kc < 4; ++kc) {
      v2f bk = *(const v2f*)(kp + 4 * kc + 2 * kh);  // B[k][n], B[k+1][n]
      s = wmma_f32_16x16x4(aq[kc], bk, s);
    }
    // scale + causal mask (query = qt*16+m, key = j*16+n)
#pragma unroll
    for (int v = 0; v < 8; ++v) {
      float sv = s[v] * 0.25f;  // 1/sqrt(16)
      if (j == qt && nl > v + mrow) sv = -INFINITY;
      s[v] = sv;
    }
    // online softmax (rows live across 16-lane halves -> butterfly 1,2,4,8)
#pragma unroll
    for (int v = 0; v < 8; ++v) {
      float tmax = s[v];
#pragma unroll
      for (int msk = 1; msk < 16; msk <<= 1)
        tmax = fmaxf(tmax, __shfl_xor(tmax, msk, 32));
      float nm = fmaxf(rm[v], tmax);
      float alpha = __expf(rm[v] - nm);
      float p = __expf(s[v] - nm);
      float psum = p;
#pragma unroll
      for (int msk = 1; msk < 16; msk <<= 1) psum += __shfl_xor(psum, msk, 32);
      rl[v] = rl[v] * alpha + psum;
      o[v] *= alpha;
      rm[v] = nm;
      s[v] = p;
    }
    // re-layout P: C-layout -> LDS -> A-layout
#pragma unroll
    for (int v = 0; v < 8; ++v) Pt[v + mrow][nl] = s[v];
    __syncthreads();
    const float* vp = V + (base + j * 16) * RANK + head * D_HEAD + nl;
#pragma unroll
    for (int kc = 0; kc < 4; ++kc) {
      v2f ap;
      ap.x = Pt[nl][4 * kc + 2 * kh];
      ap.y = Pt[nl][4 * kc + 2 * kh + 1];
      v2f bv;
      bv.x = vp[(size_t)(4 * kc + 2 * kh) * RANK];
      bv.y = vp[(size_t)(4 * kc + 2 * kh + 1) * RANK];
      o = wmma_f32_16x16x4(ap, bv, o);
    }
    __syncthreads();
  }
  float* op = O + (base + qt * 16) * RANK + head * D_HEAD + nl;
#pragma unroll
  for (int v = 0; v < 8; ++v) op[(size_t)(v + mrow) * RANK] = o[v] / rl[v];
}

// ---------------- memory scores GEMM (WMMA f32, K=128) fused with top-8.
// grid = (NTOK/16, NSPLIT): split N_KNOW 8 ways for 8x more waves on the
// FLOP-dominant kernel; each split emits per-row top-8 candidates.
__global__ void memscore_kernel(const float* __restrict__ Qm,  // [NTOK,128]
                                const float* __restrict__ kK,  // [N_KNOW,128]
                                float* __restrict__ candS,     // [NTOK,NSPLIT,8]
                                int* __restrict__ candI) {     // [NTOK,NSPLIT,8]
  int tb = blockIdx.x;      // 16-token tile
  int split = blockIdx.y;   // column partition
  int tid = threadIdx.x;    // 256 threads = 8 waves
  int lane = tid & 31;
  int wave = tid >> 5;
  int nl = lane & 15;
  int kh = lane >> 4;
  int mrow = kh ? 8 : 0;
  size_t tokBase = (size_t)tb * 16;

  __shared__ float sc[16][CHUNK + 4];  // pad: 516 % 64 banks = 4 -> conflict free
  __shared__ float cs[16][128];
  __shared__ int ci[16][128];

  // A fragments for the Q tile: K=128 -> 32 chunks, kept in registers
  const float* qp = Qm + (tokBase + nl) * RANK;
  v2f aq[32];
#pragma unroll
  for (int kc = 0; kc < 32; ++kc) aq[kc] = *(const v2f*)(qp + 4 * kc + 2 * kh);

  float ts[8];
  int ti8[8];
#pragma unroll
  for (int i = 0; i < 8; ++i) { ts[i] = -INFINITY; ti8[i] = 0; }
  int row = tid & 15, slot = tid >> 4;
  const float scale = 0.08838834764831845f;  // 1/sqrt(128)

  for (int ch = split * CHPS; ch < (split + 1) * CHPS; ++ch) {
    // stage A: 8 waves x 4 n-tiles of 16 cols = 512 cols via WMMA
#pragma unroll 1
    for (int t = 0; t < 4; ++t) {
      int ntl = wave * 4 + t;
      int col0 = ch * CHUNK + ntl * 16;
      const float* kp = kK + (size_t)(col0 + nl) * RANK;
      // prefetch the same tile of the NEXT chunk (stream is fully predictable);
      // wrap so the last chunk never targets unmapped pages
      {
        int pcol = ((col0 + CHUNK) & (N_KNOW - 1)) + nl;
        const char* pp = (const char*)(kK + (size_t)pcol * RANK) + kh * 256;
        __builtin_prefetch(pp, 0, 3);
        __builtin_prefetch(pp + 128, 0, 3);
      }
      v8f c;
#pragma unroll
      for (int v = 0; v < 8; ++v) c[v] = 0.f;
#pragma unroll
      for (int kc = 0; kc < 32; ++kc) {
        v2f bk = *(const v2f*)(kp + 4 * kc + 2 * kh);
        c = wmma_f32_16x16x4(aq[kc], bk, c);
      }
#pragma unroll
      for (int v = 0; v < 8; ++v) sc[v + mrow][ntl * 16 + nl] = c[v] * scale;
    }
    __syncthreads();
    // stage B: 16 threads per row keep a private sorted top-8
    int cbase = slot * 32;
    for (int i = 0; i < 32; ++i) {
      float s = sc[row][cbase + i];
      if (s > ts[7]) {
        ts[7] = s;
        ti8[7] = ch * CHUNK + cbase + i;
#pragma unroll
        for (int q = 7; q > 0; --q) {
          if (ts[q] > ts[q - 1]) {
            float tf = ts[q]; ts[q] = ts[q - 1]; ts[q - 1] = tf;
            int tt = ti8[q]; ti8[q] = ti8[q - 1]; ti8[q - 1] = tt;
          }
        }
      }
    }
    __syncthreads();
  }
  // merge 16 candidate lists per row -> this split's top-8 candidates
#pragma unroll
  for (int i = 0; i < 8; ++i) { cs[row][slot * 8 + i] = ts[i]; ci[row][slot * 8 + i] = ti8[i]; }
  __syncthreads();
  if (tid < 16) {
    int r = tid;
    size_t ob = ((tokBase + r) * NSPLIT + split) * 8;
    for (int k = 0; k < 8; ++k) {
      float m = -INFINITY;
      int mi = 0;
      for (int c2 = 0; c2 < 128; ++c2)
        if (cs[r][c2] > m) { m = cs[r][c2]; mi = c2; }
      candS[ob + k] = m;
      candI[ob + k] = ci[r][mi];
      cs[r][mi] = -INFINITY;
    }
  }
}

// ---- merge NSPLIT candidate lists (disjoint column ranges -> exact top-8)
__global__ void memmerge_kernel(const float* __restrict__ candS,
                                const int* __restrict__ candI,
                                int* __restrict__ outIdx, float* __restrict__ outW) {
  int tok = blockIdx.x * blockDim.x + threadIdx.x;  // 1024 tokens, 128/block
  if (tok >= NTOK) return;
  const float* s = candS + (size_t)tok * NCAND;
  const int* ix = candI + (size_t)tok * NCAND;
  unsigned long long used = 0ull;
  float best[KNOW_K];
  int bi[KNOW_K];
  for (int k = 0; k < KNOW_K; ++k) {
    float m = -INFINITY;
    int mi = 0;
    for (int c = 0; c < NCAND; ++c) {
      if (!((used >> c) & 1ull)) {
        float v = s[c];
        if (v > m) { m = v; mi = c; }
      }
    }
    used |= 1ull << mi;
    best[k] = m;
    bi[k] = ix[mi];
  }
  float mx = best[0], sum = 0.f, w[KNOW_K];
  for (int k = 0; k < KNOW_K; ++k) { w[k] = expf(best[k] - mx); sum += w[k]; }
  float inv = 1.f / sum;
  size_t ob = (size_t)tok * KNOW_K;
  for (int k = 0; k < KNOW_K; ++k) { outIdx[ob + k] = bi[k]; outW[ob + k] = w[k] * inv; }
}

// ------------------------------------------- weighted gather of kV + residual
__global__ void memgather_kernel(const int* __restrict__ idx, const float* __restrict__ w,
                                 const float* __restrict__ kV, float* __restrict__ x) {
  int tok = blockIdx.x;
  int tid = threadIdx.x;  // 128
  __shared__ int si[8];
  __shared__ float swt[8];
  if (tid < 8) { si[tid] = idx[(size_t)tok * 8 + tid]; swt[tid] = w[(size_t)tok * 8 + tid]; }
  __syncthreads();
  for (int d = tid; d < D_MODEL; d += 128) {
    float acc = 0.f;
#pragma unroll
    for (int k = 0; k < 8; ++k) acc += swt[k] * kV[(size_t)si[k] * D_MODEL + d];
    x[(size_t)tok * D_MODEL + d] += acc;
  }
}

extern "C" void kernel_launch(void* const* d_in, const int* in_sizes, int n_in,
                              void* d_out, int out_size, void* d_ws, size_t ws_size,
                              hipStream_t stream) {
  const float* x_in = (const float*)d_in[0];
  const float* cneur = (const float*)d_in[1];
  const float* eneur = (const float*)d_in[2];
  const float* kK = (const float*)d_in[3];
  const float* kV = (const float*)d_in[4];
  const float* rq = (const float*)d_in[5];
  const float* rk = (const float*)d_in[6];
  const float* rv = (const float*)d_in[7];
  const float* ro = (const float*)d_in[8];
  const float* rm = (const float*)d_in[9];
  const float* ln1g = (const float*)d_in[10];
  const float* ln1b = (const float*)d_in[11];
  const float* ln2g = (const float*)d_in[12];
  const float* ln2b = (const float*)d_in[13];
  const float* lnfg = (const float*)d_in[14];
  const float* lnfb = (const float*)d_in[15];
  float* out = (float*)d_out;

  float* ws = (float*)d_ws;
  float* xb = ws;     ws += (size_t)NTOK * D_MODEL;
  float* h = ws;      ws += (size_t)NTOK * D_MODEL;
  float* qc = ws;     ws += (size_t)NTOK * RANK;
  float* kc = ws;     ws += (size_t)NTOK * RANK;
  float* vc = ws;     ws += (size_t)NTOK * RANK;
  float* ctx = ws;    ws += (size_t)NTOK * RANK;
  float* qm = ws;     ws += (size_t)NTOK * RANK;
  float* wt = ws;     ws += (size_t)NTOK * KNOW_K;
  float* candS = ws;  ws += (size_t)NTOK * NCAND;
  int* candI = (int*)ws;  ws += (size_t)NTOK * NCAND;
  int* idx = (int*)ws;

  hipMemcpyAsync(xb, x_in, sizeof(float) * (size_t)NTOK * D_MODEL,
                 hipMemcpyDeviceToDevice, stream);

  for (int l = 0; l < N_LAYERS; ++l) {
    ln_kernel<<<NTOK, 256, 0, stream>>>(xb, ln1g + l * D_MODEL, ln1b + l * D_MODEL, h);
    compress_kernel<<<NTOK, 128, 0, stream>>>(h, rq + (size_t)l * N_COMPRESS * D_MODEL, cneur, qc);
    compress_kernel<<<NTOK, 128, 0, stream>>>(h, rk + (size_t)l * N_COMPRESS * D_MODEL, cneur, kc);
    compress_kernel<<<NTOK, 128, 0, stream>>>(h, rv + (size_t)l * N_COMPRESS * D_MODEL, cneur, vc);
    attn_kernel<<<BATCH * N_HEADS * (SEQ / 16), 32, 0, stream>>>(qc, kc, vc, ctx);
    expand_kernel<<<NTOK, 256, 0, stream>>>(ctx, ro + (size_t)l * N_EXPAND * RANK, eneur, xb);
    ln_kernel<<<NTOK, 256, 0, stream>>>(xb, ln2g + l * D_MODEL, ln2b + l * D_MODEL, h);
    compress_kernel<<<NTOK, 128, 0, stream>>>(h, rm + (size_t)l * N_COMPRESS * D_MODEL, cneur, qm);
    memscore_kernel<<<dim3(NTOK / 16, NSPLIT), 256, 0, stream>>>(qm, kK, candS, candI);
    memmerge_kernel<<<NTOK / 128, 128, 0, stream>>>(candS, candI, idx, wt);
    memgather_kernel<<<NTOK, 128, 0, stream>>>(idx, wt, kV, xb);
  }
  ln_kernel<<<NTOK, 256, 0, stream>>>(xb, lnfg, lnfb, out);
}